// SymmetrizedBFAN_38912403702331
// MI455X (gfx1250) — compile-verified
//
#include <hip/hip_runtime.h>
#include <hip/hip_bf16.h>
#include <stdint.h>

// SymmetrizedBFAN forward for MI455X (gfx1250).
// Memory-bound (48 MB traffic -> ~2.1 us floor @ 23.3 TB/s). Readout collapsed
// to piecewise-linear prefix-sum LUT; x staged via TDM (tensor_load_to_lds)
// with double buffering + s_wait_tensorcnt.

typedef unsigned int v4u __attribute__((ext_vector_type(4)));
typedef int          v8i __attribute__((ext_vector_type(8)));
typedef int          v4i __attribute__((ext_vector_type(4)));
typedef float        v2f __attribute__((ext_vector_type(2)));

#define HH   4             // mixing heads
#define DD   5             // input dim
#define KK   20            // knots
#define NF   8             // 2*H mixed features (hr0..3, hi0..3)
#define TS   256           // samples per tile == blockDim.x
#define TFL  (TS * 2 * DD) // floats per tile = 2560 (10 KB)

// Build Tensor DMA Descriptor (ISA cdna5 §8) for a contiguous 1-row tile of
// TFL dwords: tile_dim0 = TFL, tensor_dim0 = remaining dwords (OOB -> zero).
__device__ __forceinline__ void tdm_load_tile(unsigned lds_off,
                                              const float* gptr,
                                              unsigned rem_dwords) {
  unsigned long long ga = (unsigned long long)(uintptr_t)gptr;
  v4u g0;
  g0[0] = 1u;                                   // count=1, user descriptor
  g0[1] = lds_off;                              // LDS byte address
  g0[2] = (unsigned)ga;                         // global_addr[31:0]
  g0[3] = (unsigned)(ga >> 32) | 0x80000000u;   // global_addr[56:32] | type=2
  v8i g1;
  g1[0] = (int)(2u << 16);                      // data_size=4B, no mask/pad
  g1[1] = (int)((rem_dwords & 0xFFFFu) << 16);  // tensor_dim0[15:0]
  g1[2] = (int)((rem_dwords >> 16) | (1u << 16)); // tensor_dim0[31:16] | tensor_dim1=1
  g1[3] = (int)((unsigned)TFL << 16);           // tile_dim0 = 2560
  g1[4] = 1;                                    // tile_dim1=1, tile_dim2=0
  g1[5] = TFL;                                  // tensor_dim0_stride lo
  g1[6] = 0;                                    // stride hi / dim1_stride lo
  g1[7] = 0;
  v4i g2 = {0, 0, 0, 0};
  v4i g3 = {0, 0, 0, 0};
  v8i g4 = {0, 0, 0, 0, 0, 0, 0, 0};            // 6-arg toolchain form
  __builtin_amdgcn_tensor_load_to_lds(g0, g1, g2, g3, g4, 0);
}

__global__ __launch_bounds__(TS) void SymmetrizedBFAN_38912403702331_kernel(
    const float* __restrict__ x,        // [B, D, 2] interleaved (re, im)
    const float* __restrict__ mwre,     // [H, D]
    const float* __restrict__ mwim,     // [H, D]
    const float* __restrict__ rore,     // [1, 160] = [NF, KK]
    const float* __restrict__ roim,     // [1, 160]
    float* __restrict__ out,            // [B, 1, 2]
    int B, int num_tiles)
{
  __shared__ float4 s_tab[NF * 21];                 // {SAre, SBre, SAim, SBim}
  __shared__ __align__(16) float s_x[2][TFL];       // TDM double buffer

  const int tid = threadIdx.x;

  // ---- one-time per-block: prefix-sum tables for piecewise-linear readout.
  // sum_k w_k * max(|h| - c_k, 0) over active knots j = #{c_k < |h|}
  //   = SA[j] * |h| - SB[j],  SA[j] = sum_{k<j} w_k,  SB[j] = sum_{k<j} w_k*c_k
  for (int idx = tid; idx < NF * 21; idx += TS) {
    int f = idx / 21, j = idx % 21;
    float sar = 0.f, sbr = 0.f, sai = 0.f, sbi = 0.f;
    for (int k = 0; k < j; ++k) {
      float c  = 0.2f * (float)k / 19.0f;           // linspace(0,0.2,20)
      float wr = rore[f * KK + k];
      float wi = roim[f * KK + k];
      sar += wr; sbr += wr * c;
      sai += wi; sbi += wi * c;
    }
    s_tab[idx] = make_float4(sar, sbr, sai, sbi);
  }

  // ---- mixing weights: uniform addresses -> scalar loads -> SGPRs
  float wre[HH * DD], wim[HH * DD];
#pragma unroll
  for (int i = 0; i < HH * DD; ++i) { wre[i] = mwre[i]; wim[i] = mwim[i]; }

  const unsigned total_dw = (unsigned)B * (unsigned)(2 * DD);

  // ---- prologue: wave 0 issues TDM for this block's first tile
  int tile = blockIdx.x;
  int buf = 0;
  if (tile < num_tiles && tid < 32) {
    tdm_load_tile((unsigned)(uintptr_t)&s_x[0][0],
                  x + (size_t)tile * TFL,
                  total_dw - (unsigned)tile * TFL);
  }
  __syncthreads();  // covers table build; data readiness gated per-iter below

  for (; tile < num_tiles; tile += gridDim.x) {
    int nxt = tile + (int)gridDim.x;
    if (tid < 32) {  // wave 0 drives the TDM pipeline
      if (nxt < num_tiles) {
        tdm_load_tile((unsigned)(uintptr_t)&s_x[buf ^ 1][0],
                      x + (size_t)nxt * TFL,
                      total_dw - (unsigned)nxt * TFL);
        __builtin_amdgcn_s_wait_tensorcnt(1);  // oldest (current tile) done
      } else {
        __builtin_amdgcn_s_wait_tensorcnt(0);  // drain
      }
    }
    __syncthreads();  // current tile visible to all waves

    long long s = (long long)tile * TS + tid;
    if (s < B) {
      // launder LDS pointer: TDM writes are invisible to the compiler
      const float* xbuf = &s_x[buf][0];
      asm volatile("" : "+v"(xbuf) :: "memory");

      const v2f* xp = (const v2f*)(xbuf + tid * (2 * DD));
      v2f p0 = xp[0], p1 = xp[1], p2 = xp[2], p3 = xp[3], p4 = xp[4];
      float xr[DD] = {p0.x, p1.x, p2.x, p3.x, p4.x};
      float xi[DD] = {p0.y, p1.y, p2.y, p3.y, p4.y};

      // complex mix: hr = xr*Wre^T - xi*Wim^T ; hi = xr*Wim^T + xi*Wre^T
      float hv[NF];
#pragma unroll
      for (int h = 0; h < HH; ++h) {
        float hr = 0.f, hi = 0.f;
#pragma unroll
        for (int d = 0; d < DD; ++d) {
          hr = fmaf(xr[d],  wre[h * DD + d], hr);
          hr = fmaf(-xi[d], wim[h * DD + d], hr);
          hi = fmaf(xr[d],  wim[h * DD + d], hi);
          hi = fmaf(xi[d],  wre[h * DD + d], hi);
        }
        hv[h] = hr;
        hv[HH + h] = hi;
      }

      // readout via LUT: contrib = SA*h - sign(h)*SB  (= s*(SA*|h| - SB))
      float accr = 0.f, acci = 0.f;
#pragma unroll
      for (int f = 0; f < NF; ++f) {
        float h = hv[f];
        unsigned hb  = __float_as_uint(h);
        unsigned sgn = hb & 0x80000000u;
        float a = __uint_as_float(hb & 0x7FFFFFFFu);      // |h|
        int j = (int)(a * 95.0f);                          // 95 = 19/0.2
        j = j < 19 ? j : 19;
        float4 tb = s_tab[f * 21 + j + 1];                 // ds_load_b128
        accr  = fmaf(tb.x, h, accr);
        accr -= __uint_as_float(__float_as_uint(tb.y) ^ sgn);
        acci  = fmaf(tb.z, h, acci);
        acci -= __uint_as_float(__float_as_uint(tb.w) ^ sgn);
      }

      v2f res; res.x = accr; res.y = acci;
      __builtin_nontemporal_store(res, (v2f*)out + s);
    }
    __syncthreads();  // buf must be fully consumed before it becomes a TDM dest
    buf ^= 1;
  }
}

extern "C" void kernel_launch(void* const* d_in, const int* in_sizes, int n_in,
                              void* d_out, int out_size, void* d_ws, size_t ws_size,
                              hipStream_t stream) {
  (void)n_in; (void)out_size; (void)d_ws; (void)ws_size;
  const float* x    = (const float*)d_in[0];
  const float* mwre = (const float*)d_in[1];
  const float* mwim = (const float*)d_in[2];
  const float* rore = (const float*)d_in[3];
  const float* roim = (const float*)d_in[4];
  float* out = (float*)d_out;

  int B = in_sizes[0] / (2 * DD);                 // 1048576
  int num_tiles = (B + TS - 1) / TS;              // 4096
  int grid = num_tiles < 1024 ? num_tiles : 1024; // persistent WGs, 4 tiles each

  SymmetrizedBFAN_38912403702331_kernel<<<grid, TS, 0, stream>>>(
      x, mwre, mwim, rore, roim, out, B, num_tiles);
}